// AdaptUnsupFocalLoss_86681029968183
// MI455X (gfx1250) — compile-verified
//
#include <hip/hip_runtime.h>
#include <hip/hip_bf16.h>
#include <math.h>

// Problem geometry (fixed by the reference): N=2, C=19, H=512, W=1024
#define NCLS 19
#define HW_SIZE (512 * 1024)          // 524288 = 2^19
#define NPIX (2 * HW_SIZE)            // 1048576 pixels
#define CH_STRIDE ((size_t)HW_SIZE)
#define BATCH_STRIDE ((size_t)NCLS * (size_t)HW_SIZE)
#define NBINS 65536
#define FIX_SCALE 16777216.0f         // 2^24 fixed-point for deterministic loss sum
#define TILE_PIX 32                   // pixels per wave tile (one wave32)
#define WAVES_PER_BLOCK 8
#define MAIN_BLOCKS 2048              // 2048*256 = 524288 threads -> 2 uniform iters

typedef __attribute__((ext_vector_type(2))) float v2f;
typedef __attribute__((ext_vector_type(8))) float v8f;
typedef unsigned int uintx4 __attribute__((ext_vector_type(4)));
typedef int intx4 __attribute__((ext_vector_type(4)));
typedef int intx8 __attribute__((ext_vector_type(8)));

#if __has_builtin(__builtin_amdgcn_tensor_load_to_lds)
#define TDM_OK 1
#else
#define TDM_OK 0
#endif

struct Small {
  unsigned long long loss_fixed;      // fixed-point sum of masked loss
  unsigned int solid_count;
  unsigned int mask_count;
  unsigned int bucket[NCLS];          // level-1 selected top-16-bit bucket per class
  unsigned int rank[NCLS];            // 0-based descending rank within bucket
  float nthr[NCLS];                   // new per-class thresholds
  unsigned int pad;
};

// Workspace layout (bytes). Total ~14.42 MB (L2-resident: 192 MB L2).
#define HIST_OFF  ((size_t)0)                         // u32[19*65536] = 4,980,736 B
#define CONF_OFF  ((size_t)4980736)                   // f32[NPIX]     = 4 MB
#define LOSS_OFF  ((size_t)9175040)                   // f32[NPIX]     = 4 MB
#define LABEL_OFF ((size_t)13369344)                  // u8[NPIX]      = 1 MB
#define SMALL_OFF ((size_t)14417920)                  // struct Small

// ---------------------------------------------------------------------------
// Exact f32 wave(32) sum on the matrix unit (V_WMMA_F32_16X16X4_F32):
// A(16x4): lane L<16 -> A[L][0]=x ; lane L>=16 -> A[L-16][2]=x ; B = ones
// => D[m][n] = x_m + x_{m+16}; each lane sums its 8 acc VGPRs, one half-wave
// swap finishes. Requires EXEC == all ones (call sites have uniform trips).
// ---------------------------------------------------------------------------
static __device__ __forceinline__ float wave_reduce_sum_wmma(float x) {
  v2f a; a[0] = x;    a[1] = 0.0f;
  v2f b; b[0] = 1.0f; b[1] = 1.0f;
  v8f acc = {};
  acc = __builtin_amdgcn_wmma_f32_16x16x4_f32(false, a, false, b,
                                              (short)0, acc, false, false);
  float s = acc[0] + acc[1] + acc[2] + acc[3] + acc[4] + acc[5] + acc[6] + acc[7];
  s += __shfl_xor(s, 16, 32);
  return s;
}

// LDS byte offset of a __shared__ object (addrspacecast generic -> local).
static __device__ __forceinline__ unsigned lds_off(const void* p) {
  return (unsigned)(uintptr_t)(__attribute__((address_space(3))) const char*)p;
}

// ---------------------------------------------------------------------------
// Stage one wave tile (19 channels x 32 pixels, channel stride = HW) into LDS.
// TDM path: one tensor_load_to_lds descriptor per tensor per wave.
// ---------------------------------------------------------------------------
static __device__ __forceinline__ void stage_tile(const float* __restrict__ g,
                                                  float (*t)[TILE_PIX], int lane) {
#if TDM_OK
  unsigned glo = __builtin_amdgcn_readfirstlane((unsigned)(uintptr_t)g);
  unsigned ghi = __builtin_amdgcn_readfirstlane((unsigned)((uintptr_t)g >> 32));
  unsigned laddr = __builtin_amdgcn_readfirstlane(lds_off(&t[0][0]));
  // D# group0: count=1 | lds_addr | global_addr[56:0] | type=2
  uintx4 g0;
  g0[0] = 1u;
  g0[1] = laddr;
  g0[2] = glo;
  g0[3] = (ghi & 0x01FFFFFFu) | 0x80000000u;
  // D# group1: data_size=4B; tensor_dim0=HW; tensor_dim1=19; tile=32x19;
  //            tensor_dim0_stride=HW (2D tile: tile_dim2=0, dim1_stride unused)
  intx8 g1;
  g1[0] = 0x00020000;                      // data_size=2 (4 bytes)
  g1[1] = 0;                               // dim0 low16 = 0 (HW = 0x80000)
  g1[2] = (int)((19u << 16) | 0x8u);       // dim0 hi16 = 8, tensor_dim1 lo16 = 19
  g1[3] = (int)(32u << 16);                // tensor_dim1 hi = 0, tile_dim0 = 32
  g1[4] = 19;                              // tile_dim1 = 19, tile_dim2 = 0
  g1[5] = (int)0x80000u;                   // tensor_dim0_stride[31:0] = HW
  g1[6] = 0;
  g1[7] = 0;
  intx4 z4 = {0, 0, 0, 0};
#if defined(__clang_major__) && (__clang_major__ >= 23)
  intx8 z8 = {0, 0, 0, 0, 0, 0, 0, 0};
  __builtin_amdgcn_tensor_load_to_lds(g0, g1, z4, z4, z8, 0);
#else
  __builtin_amdgcn_tensor_load_to_lds(g0, g1, z4, z4, 0);
#endif
#else
#pragma unroll
  for (int c = 0; c < NCLS; ++c)
    t[c][lane] = g[(size_t)c * CH_STRIDE + lane];
#endif
}

static __device__ __forceinline__ void tile_fence() {
#if TDM_OK
  __builtin_amdgcn_s_wait_tensorcnt(0);
#endif
  __asm__ volatile("" ::: "memory");   // DMA wrote LDS behind the compiler's back
}

// ---------------------------------------------------------------------------
__global__ void init_kernel(unsigned int* hist, Small* sm) {
  int tid = blockIdx.x * blockDim.x + threadIdx.x;
  int stride = gridDim.x * blockDim.x;
  const int nwords = NCLS * NBINS;
  for (int i = tid; i < nwords; i += stride) hist[i] = 0u;
  if (blockIdx.x == 0 && threadIdx.x < (int)(sizeof(Small) / 4))
    ((unsigned int*)sm)[threadIdx.x] = 0u;
}

// ---------------------------------------------------------------------------
// Main fused pass: TDM-staged tiles -> softmax stats, label/conf/loss,
// solid count (WMMA wave sum), level-1 radix histogram.
// loss = (1 - pa[label]) * (log(sum_ea) + ma - sum_c pb[c]*la[c])
// ---------------------------------------------------------------------------
__global__ void main_kernel(const float* __restrict__ lbp,
                            const float* __restrict__ lap,
                            float* __restrict__ conf, float* __restrict__ loss,
                            unsigned char* __restrict__ label,
                            unsigned int* __restrict__ hist, Small* sm) {
  __shared__ float tile[WAVES_PER_BLOCK][2][NCLS][TILE_PIX];   // 38,912 B
  const int lane = threadIdx.x & 31;
  const int wid = threadIdx.x >> 5;
  int tid = blockIdx.x * blockDim.x + threadIdx.x;
  int stride = gridDim.x * blockDim.x;            // == HW_SIZE
  float solid_local = 0.0f;

  for (int p = tid; p < NPIX; p += stride) {
    int pbase = p - lane;                         // wave's first pixel
    size_t gb = ((size_t)(pbase >> 19)) * BATCH_STRIDE +
                (size_t)(pbase & (HW_SIZE - 1));
    stage_tile(lbp + gb, tile[wid][0], lane);
    stage_tile(lap + gb, tile[wid][1], lane);
    tile_fence();

    float mb = -1e30f, ma = -1e30f;
    int lab = 0;
#pragma unroll
    for (int c = 0; c < NCLS; ++c) {
      float x = tile[wid][0][c][lane];
      float y = tile[wid][1][c][lane];
      if (x > mb) { mb = x; lab = c; }            // first-max like jnp.argmax
      ma = fmaxf(ma, y);
    }
    float seb = 0.0f, sea = 0.0f, dot = 0.0f, ea_lab = 0.0f;
#pragma unroll
    for (int c = 0; c < NCLS; ++c) {
      float x = tile[wid][0][c][lane];
      float y = tile[wid][1][c][lane];
      float eb = __expf(x - mb);
      float ea = __expf(y - ma);
      seb += eb;
      dot += eb * y;
      sea += ea;
      if (c == lab) ea_lab = ea;
    }

    float confv = 1.0f / seb;                     // max softmax prob
    float pa_lab = ea_lab / sea;                  // pa at argmax(pb)
    float lossv = (1.0f - pa_lab) * (__logf(sea) + ma - dot / seb);

    conf[p] = confv;
    loss[p] = lossv;
    label[p] = (unsigned char)lab;
    solid_local += (confv > 0.8f) ? 1.0f : 0.0f;

    unsigned int bits = __float_as_uint(confv);   // conf>0: uint order==float order
    atomicAdd(&hist[(unsigned)lab * NBINS + (bits >> 16)], 1u);
  }

  float wsum = wave_reduce_sum_wmma(solid_local); // uniform trips -> EXEC all 1s
  if ((threadIdx.x & 31) == 0)
    atomicAdd(&sm->solid_count, (unsigned int)(wsum + 0.5f));
}

// ---------------------------------------------------------------------------
// Level-1 select: per class, bucket (top-16 bits) + in-bucket rank of the
// idx-th largest of {class-c confs} U {thr}; then zero hist for level 2.
// ---------------------------------------------------------------------------
__global__ void select1_kernel(const float* __restrict__ cls_thresh,
                               unsigned int* __restrict__ hist, Small* sm) {
  int c = blockIdx.x;
  unsigned int* h = hist + (unsigned)c * NBINS;
  __shared__ unsigned int chunk[256];
  int t = threadIdx.x;

  unsigned int s = 0;
  for (int i = 0; i < 256; ++i) s += h[t * 256 + i];
  chunk[t] = s;
  __syncthreads();

  if (t == 0) {
    float thr = cls_thresh[c];
    unsigned int tb = __float_as_uint(thr);
    unsigned int tbucket = tb >> 16;
    unsigned int tchunk = tbucket >> 8;

    unsigned int count = 0;
    for (int i = 0; i < 256; ++i) count += chunk[i];
    unsigned int idx =
        (unsigned int)floorf((float)(count + 1u) * 0.2f * __powf(thr, 8.0f));

    unsigned int cum = 0;
    int ck = 255;
    for (; ck >= 0; --ck) {
      unsigned int cc = chunk[ck] + (((unsigned)ck == tchunk) ? 1u : 0u);
      if (cum + cc > idx) break;
      cum += cc;
    }
    if (ck < 0) ck = 0;

    unsigned int bsel = (unsigned)ck * 256u, r = 0;
    for (int b = ck * 256 + 255; b >= ck * 256; --b) {
      unsigned int cc = h[b] + (((unsigned)b == tbucket) ? 1u : 0u);
      if (cum + cc > idx) { bsel = (unsigned)b; r = idx - cum; break; }
      cum += cc;
    }
    sm->bucket[c] = bsel;
    sm->rank[c] = r;
  }
  __syncthreads();
  for (int i = 0; i < 256; ++i) h[t * 256 + i] = 0u;   // reuse as level-2 hist
}

// ---------------------------------------------------------------------------
// Level-2 histogram over low-16 conf bits within each class's selected bucket.
// ---------------------------------------------------------------------------
__global__ void hist2_kernel(const float* __restrict__ conf,
                             const unsigned char* __restrict__ label,
                             unsigned int* __restrict__ hist, const Small* sm) {
  __shared__ unsigned int sb[NCLS];
  if (threadIdx.x < NCLS) sb[threadIdx.x] = sm->bucket[threadIdx.x];
  __syncthreads();
  int tid = blockIdx.x * blockDim.x + threadIdx.x;
  int stride = gridDim.x * blockDim.x;
  const int nq = NPIX / 4;
  const float4* c4 = (const float4*)conf;
  const uchar4* l4 = (const uchar4*)label;
  for (int q = tid; q < nq; q += stride) {
    if (q + stride < nq) __builtin_prefetch(&c4[q + stride], 0, 3);
    float4 cv = c4[q];
    uchar4 lv = l4[q];
    unsigned int b0 = __float_as_uint(cv.x), b1 = __float_as_uint(cv.y);
    unsigned int b2 = __float_as_uint(cv.z), b3 = __float_as_uint(cv.w);
    if ((b0 >> 16) == sb[lv.x]) atomicAdd(&hist[(unsigned)lv.x * NBINS + (b0 & 0xFFFFu)], 1u);
    if ((b1 >> 16) == sb[lv.y]) atomicAdd(&hist[(unsigned)lv.y * NBINS + (b1 & 0xFFFFu)], 1u);
    if ((b2 >> 16) == sb[lv.z]) atomicAdd(&hist[(unsigned)lv.z * NBINS + (b2 & 0xFFFFu)], 1u);
    if ((b3 >> 16) == sb[lv.w]) atomicAdd(&hist[(unsigned)lv.w * NBINS + (b3 & 0xFFFFu)], 1u);
  }
}

// ---------------------------------------------------------------------------
// Level-2 select + adaptive threshold update.
// ---------------------------------------------------------------------------
__global__ void select2_kernel(const float* __restrict__ cls_thresh,
                               const unsigned int* __restrict__ hist, Small* sm) {
  int c = blockIdx.x;
  const unsigned int* h = hist + (unsigned)c * NBINS;
  __shared__ unsigned int chunk[256];
  int t = threadIdx.x;

  unsigned int s = 0;
  for (int i = 0; i < 256; ++i) s += h[t * 256 + i];
  chunk[t] = s;
  __syncthreads();

  if (t == 0) {
    float thr = cls_thresh[c];
    unsigned int tb = __float_as_uint(thr);
    unsigned int tbucket = tb >> 16, tlow = tb & 0xFFFFu;
    unsigned int B = sm->bucket[c], r = sm->rank[c];
    bool thr_here = (tbucket == B);
    unsigned int tch = tlow >> 8;

    unsigned int cum = 0;
    int ck = 255;
    for (; ck >= 0; --ck) {
      unsigned int cc = chunk[ck] + ((thr_here && (unsigned)ck == tch) ? 1u : 0u);
      if (cum + cc > r) break;
      cum += cc;
    }
    if (ck < 0) ck = 0;

    unsigned int lsel = (unsigned)ck * 256u;
    for (int b = ck * 256 + 255; b >= ck * 256; --b) {
      unsigned int cc = h[b] + ((thr_here && (unsigned)b == tlow) ? 1u : 0u);
      if (cum + cc > r) { lsel = (unsigned)b; break; }
      cum += cc;
    }

    float tmp = __uint_as_float((B << 16) | lsel);   // exact order statistic
    float nt = 0.9f * thr + 0.1f * tmp;              // ADAPT_A = 0.9
    if (nt >= 1.0f) nt = 0.999f;
    sm->nthr[c] = nt;
  }
}

// ---------------------------------------------------------------------------
// Final masked reduction (vectorized; WMMA wave sums + deterministic atomics).
// ---------------------------------------------------------------------------
__global__ void final_kernel(const float* __restrict__ conf,
                             const float* __restrict__ loss,
                             const unsigned char* __restrict__ label, Small* sm) {
  __shared__ float nthr[NCLS];
  if (threadIdx.x < NCLS) nthr[threadIdx.x] = sm->nthr[threadIdx.x];
  __syncthreads();

  int tid = blockIdx.x * blockDim.x + threadIdx.x;
  int stride = gridDim.x * blockDim.x;
  const int nq = NPIX / 4;
  const float4* c4 = (const float4*)conf;
  const float4* s4 = (const float4*)loss;
  const uchar4* l4 = (const uchar4*)label;
  float cnt = 0.0f, ls = 0.0f;
  for (int q = tid; q < nq; q += stride) {
    if (q + stride < nq) {
      __builtin_prefetch(&c4[q + stride], 0, 3);
      __builtin_prefetch(&s4[q + stride], 0, 3);
    }
    float4 cv = c4[q];
    float4 lo = s4[q];
    uchar4 lv = l4[q];
    if (cv.x > nthr[lv.x]) { cnt += 1.0f; ls += fmaxf(lo.x, 1e-8f); }
    if (cv.y > nthr[lv.y]) { cnt += 1.0f; ls += fmaxf(lo.y, 1e-8f); }
    if (cv.z > nthr[lv.z]) { cnt += 1.0f; ls += fmaxf(lo.z, 1e-8f); }
    if (cv.w > nthr[lv.w]) { cnt += 1.0f; ls += fmaxf(lo.w, 1e-8f); }
  }
  float wc = wave_reduce_sum_wmma(cnt);   // exact: integer-valued f32
  float wl = wave_reduce_sum_wmma(ls);
  if ((threadIdx.x & 31) == 0) {
    atomicAdd(&sm->mask_count, (unsigned int)(wc + 0.5f));
    atomicAdd(&sm->loss_fixed, (unsigned long long)(wl * FIX_SCALE));
  }
}

__global__ void writeout_kernel(const Small* sm, float* out) {
  if (blockIdx.x == 0 && threadIdx.x == 0) {
    double num = (double)sm->loss_fixed / (double)FIX_SCALE;
    float den = fmaxf((float)sm->mask_count, 1.0f);
    out[0] = (float)(num / (double)den);
    out[1] = (float)sm->mask_count / (float)NPIX;
    out[2] = (float)sm->solid_count / (float)NPIX;
  }
}

// ---------------------------------------------------------------------------
extern "C" void kernel_launch(void* const* d_in, const int* in_sizes, int n_in,
                              void* d_out, int out_size, void* d_ws, size_t ws_size,
                              hipStream_t stream) {
  (void)in_sizes; (void)n_in; (void)out_size; (void)ws_size;
  const float* lbp = (const float*)d_in[0];
  const float* lap = (const float*)d_in[1];
  const float* cth = (const float*)d_in[2];
  float* out = (float*)d_out;

  char* ws = (char*)d_ws;
  unsigned int* hist = (unsigned int*)(ws + HIST_OFF);
  float* conf = (float*)(ws + CONF_OFF);
  float* loss = (float*)(ws + LOSS_OFF);
  unsigned char* label = (unsigned char*)(ws + LABEL_OFF);
  Small* sm = (Small*)(ws + SMALL_OFF);

  init_kernel<<<1024, 256, 0, stream>>>(hist, sm);
  main_kernel<<<MAIN_BLOCKS, 256, 0, stream>>>(lbp, lap, conf, loss, label, hist, sm);
  select1_kernel<<<NCLS, 256, 0, stream>>>(cth, hist, sm);
  hist2_kernel<<<512, 256, 0, stream>>>(conf, label, hist, sm);
  select2_kernel<<<NCLS, 256, 0, stream>>>(cth, hist, sm);
  final_kernel<<<512, 256, 0, stream>>>(conf, loss, label, sm);
  writeout_kernel<<<1, 64, 0, stream>>>(sm, out);
}